// OTLayer_27324581937715
// MI455X (gfx1250) — compile-verified
//
#include <hip/hip_runtime.h>
#include <math.h>

#define NA 8192
#define NT 8192
#define DD 256
#define DH 128
#define INV_EPS 10.0f
#define LOG_A (-9.0109133f)   // log(1/8192)
#define NEG_INF (-__builtin_inff())
#define RC 64                 // row-chunks for column-LSE partials (128 rows each)

typedef __attribute__((ext_vector_type(2))) float v2f;
typedef __attribute__((ext_vector_type(8))) float v8f;

// ---------- online logsumexp helper ----------
__device__ __forceinline__ void lse_update(float& m, float& s, float v) {
    if (v > m) { s = s * __expf(m - v) + 1.0f; m = v; }
    else       { s += __expf(v - m); }
}

// ---------- generic WMMA fp32 GEMM: C[M,N] = A[M,K] @ B[K,N] + bias ----------
// one wave (32 threads) per 16x16 output tile, K-loop of V_WMMA_F32_16X16X4_F32
template <bool RELU>
__global__ void wmma_gemm_bias(const float* __restrict__ A, const float* __restrict__ B,
                               const float* __restrict__ bias, float* __restrict__ C,
                               int M, int N, int K) {
    const int lane = threadIdx.x;
    const int l16  = lane & 15;
    const int half = lane >> 4;
    const int bi = blockIdx.y, bj = blockIdx.x;
    const int arow = bi * 16 + l16;   // A operand: lane holds row M=l16 of tile
    const int bcol = bj * 16 + l16;   // B operand: lane holds col N=l16 of tile

    const float* ap = A + (size_t)arow * K + half * 2;     // K = half*2, half*2+1
    const float* bp = B + (size_t)(half * 2) * N + bcol;

    v8f acc = {};
    for (int kk = 0; kk < K; kk += 4) {
        v2f a = *reinterpret_cast<const v2f*>(ap);         // contiguous, 8B aligned
        v2f b;
        b.x = bp[0];
        b.y = bp[(size_t)N];
        acc = __builtin_amdgcn_wmma_f32_16x16x4_f32(false, a, false, b,
                                                    (short)0, acc, false, false);
        ap += 4;
        bp += (size_t)4 * N;
    }
    const float bv = bias[bcol];
#pragma unroll
    for (int r = 0; r < 8; ++r) {
        const int m = bi * 16 + half * 8 + r;              // C/D layout: VGPR r -> M
        float v = acc[r] + bv;
        if (RELU) v = fmaxf(v, 0.0f);
        C[(size_t)m * N + bcol] = v;
    }
}

// ---------- log_K = (||q||^2 + ||k||^2 - 2 q.k)/eps via WMMA ----------
// each wave computes a 16x64 strip: 4 accumulators share the A fragment
// (1 A-load + 4 B-loads per 4 WMMAs -> 1.25 b64 loads / wmma)
__global__ void logk_kernel(const float* __restrict__ q, const float* __restrict__ k,
                            const float* __restrict__ qn, const float* __restrict__ kn,
                            float* __restrict__ logK) {
    const int lane = threadIdx.x;
    const int l16  = lane & 15;
    const int half = lane >> 4;
    const int bi = blockIdx.y;        // 16-row tile of q
    const int bj = blockIdx.x;        // 64-col strip of k

    const float* ap = q + (size_t)(bi * 16 + l16) * DD + half * 2;
    const float* bp0 = k + (size_t)(bj * 64 +  0 + l16) * DD + half * 2;
    const float* bp1 = k + (size_t)(bj * 64 + 16 + l16) * DD + half * 2;
    const float* bp2 = k + (size_t)(bj * 64 + 32 + l16) * DD + half * 2;
    const float* bp3 = k + (size_t)(bj * 64 + 48 + l16) * DD + half * 2;

    v8f acc0 = {}, acc1 = {}, acc2 = {}, acc3 = {};
#pragma unroll 2
    for (int kk = 0; kk < DD; kk += 4) {
        v2f a  = *reinterpret_cast<const v2f*>(ap);
        v2f b0 = *reinterpret_cast<const v2f*>(bp0);
        v2f b1 = *reinterpret_cast<const v2f*>(bp1);
        v2f b2 = *reinterpret_cast<const v2f*>(bp2);
        v2f b3 = *reinterpret_cast<const v2f*>(bp3);
        acc0 = __builtin_amdgcn_wmma_f32_16x16x4_f32(false, a, false, b0, (short)0, acc0, false, false);
        acc1 = __builtin_amdgcn_wmma_f32_16x16x4_f32(false, a, false, b1, (short)0, acc1, false, false);
        acc2 = __builtin_amdgcn_wmma_f32_16x16x4_f32(false, a, false, b2, (short)0, acc2, false, false);
        acc3 = __builtin_amdgcn_wmma_f32_16x16x4_f32(false, a, false, b3, (short)0, acc3, false, false);
        ap += 4; bp0 += 4; bp1 += 4; bp2 += 4; bp3 += 4;
    }

    float qnv[8];
#pragma unroll
    for (int r = 0; r < 8; ++r) qnv[r] = qn[bi * 16 + half * 8 + r];

    const v8f* accs[4] = {&acc0, &acc1, &acc2, &acc3};
#pragma unroll
    for (int t = 0; t < 4; ++t) {
        const int col = bj * 64 + t * 16 + l16;
        const float knv = kn[col];
        const v8f& a = *accs[t];
#pragma unroll
        for (int r = 0; r < 8; ++r) {
            const int m = bi * 16 + half * 8 + r;
            logK[(size_t)m * NT + col] = (qnv[r] + knv - 2.0f * a[r]) * INV_EPS;
        }
    }
}

// ---------- row squared norms, 256 cols, one block per row ----------
__global__ void rownorm_kernel(const float* __restrict__ X, float* __restrict__ out) {
    const int row = blockIdx.x, tid = threadIdx.x;
    float v = X[(size_t)row * DD + tid];
    __shared__ float red[256];
    red[tid] = v * v;
    __syncthreads();
    for (int w = 128; w > 0; w >>= 1) {
        if (tid < w) red[tid] += red[tid + w];
        __syncthreads();
    }
    if (tid == 0) out[row] = red[0];
}

// ---------- supply logits: s[i] = h[i,:].W2 + b2 ----------
__global__ void logits_kernel(const float* __restrict__ h, const float* __restrict__ W2,
                              const float* __restrict__ b2, float* __restrict__ s) {
    const int row = blockIdx.x, tid = threadIdx.x;
    __shared__ float red[128];
    red[tid] = h[(size_t)row * DH + tid] * W2[tid];
    __syncthreads();
    for (int w = 64; w > 0; w >>= 1) {
        if (tid < w) red[tid] += red[tid + w];
        __syncthreads();
    }
    if (tid == 0) s[row] = red[0] + b2[0];
}

// ---------- logsumexp over the NA logits (single block) ----------
__global__ void lse_kernel(const float* __restrict__ s, float* __restrict__ lse) {
    const int tid = threadIdx.x;  // 1024
    float m = NEG_INF, ss = 0.0f;
    for (int i = tid; i < NA; i += 1024) lse_update(m, ss, s[i]);
    __shared__ float sm[1024], sv[1024];
    sm[tid] = m; sv[tid] = ss;
    __syncthreads();
    for (int w = 512; w > 0; w >>= 1) {
        if (tid < w) {
            float m1 = sm[tid], s1 = sv[tid], m2 = sm[tid + w], s2 = sv[tid + w];
            float M = fmaxf(m1, m2);
            sv[tid] = s1 * __expf(m1 - M) + s2 * __expf(m2 - M);
            sm[tid] = M;
        }
        __syncthreads();
    }
    if (tid == 0) lse[0] = sm[0] + __logf(sv[0]);
}

__global__ void logb_kernel(const float* __restrict__ s, const float* __restrict__ lse,
                            float* __restrict__ log_b) {
    const int i = blockIdx.x * 256 + threadIdx.x;
    log_b[i] = s[i] - lse[0];
}

__global__ void zero_kernel(float* __restrict__ g) {
    g[blockIdx.x * 256 + threadIdx.x] = 0.0f;
}

// ---------- f[i] = log_b[i] - LSE_j(logK[i,j] + g[j]), one block per row ----------
__global__ void f_kernel(const float* __restrict__ logK, const float* __restrict__ g,
                         const float* __restrict__ log_b, float* __restrict__ f) {
    const int i = blockIdx.x, tid = threadIdx.x;
    const float* row = logK + (size_t)i * NT;
    float m = NEG_INF, s = 0.0f;
    for (int j = tid * 4; j < NT; j += 1024) {          // float4 stream
        float4 v  = *reinterpret_cast<const float4*>(row + j);
        float4 gg = *reinterpret_cast<const float4*>(g + j);
        lse_update(m, s, v.x + gg.x);
        lse_update(m, s, v.y + gg.y);
        lse_update(m, s, v.z + gg.z);
        lse_update(m, s, v.w + gg.w);
    }
    __shared__ float sm[256], sv[256];
    sm[tid] = m; sv[tid] = s;
    __syncthreads();
    for (int w = 128; w > 0; w >>= 1) {
        if (tid < w) {
            float m1 = sm[tid], s1 = sv[tid], m2 = sm[tid + w], s2 = sv[tid + w];
            float M = fmaxf(m1, m2);
            sv[tid] = s1 * __expf(m1 - M) + s2 * __expf(m2 - M);
            sm[tid] = M;
        }
        __syncthreads();
    }
    if (tid == 0) f[i] = log_b[i] - (sm[0] + __logf(sv[0]));
}

// ---------- column LSE partials: RC row-chunks, 4 columns per thread ----------
__global__ void g_partial_kernel(const float* __restrict__ logK, const float* __restrict__ f,
                                 float* __restrict__ pm, float* __restrict__ ps) {
    const int j  = (blockIdx.x * 256 + threadIdx.x) * 4;
    const int rc = blockIdx.y;
    float4 m = {NEG_INF, NEG_INF, NEG_INF, NEG_INF};
    float4 s = {0.0f, 0.0f, 0.0f, 0.0f};
    const int i0 = rc * (NA / RC);
    for (int i = i0; i < i0 + (NA / RC); ++i) {
        float4 v = *reinterpret_cast<const float4*>(logK + (size_t)i * NT + j);
        const float fi = f[i];
        lse_update(m.x, s.x, v.x + fi);
        lse_update(m.y, s.y, v.y + fi);
        lse_update(m.z, s.z, v.z + fi);
        lse_update(m.w, s.w, v.w + fi);
    }
    *reinterpret_cast<float4*>(pm + (size_t)rc * NT + j) = m;
    *reinterpret_cast<float4*>(ps + (size_t)rc * NT + j) = s;
}

__global__ void g_combine_kernel(const float* __restrict__ pm, const float* __restrict__ ps,
                                 float* __restrict__ g) {
    const int j = (blockIdx.x * 256 + threadIdx.x) * 4;
    float4 M = {NEG_INF, NEG_INF, NEG_INF, NEG_INF};
    float4 S = {0.0f, 0.0f, 0.0f, 0.0f};
    for (int rc = 0; rc < RC; ++rc) {
        float4 m = *reinterpret_cast<const float4*>(pm + (size_t)rc * NT + j);
        float4 s = *reinterpret_cast<const float4*>(ps + (size_t)rc * NT + j);
        float n;
        n = fmaxf(M.x, m.x); S.x = S.x * __expf(M.x - n) + s.x * __expf(m.x - n); M.x = n;
        n = fmaxf(M.y, m.y); S.y = S.y * __expf(M.y - n) + s.y * __expf(m.y - n); M.y = n;
        n = fmaxf(M.z, m.z); S.z = S.z * __expf(M.z - n) + s.z * __expf(m.z - n); M.z = n;
        n = fmaxf(M.w, m.w); S.w = S.w * __expf(M.w - n) + s.w * __expf(m.w - n); M.w = n;
    }
    float4 out;
    out.x = LOG_A - (M.x + __logf(S.x));
    out.y = LOG_A - (M.y + __logf(S.y));
    out.z = LOG_A - (M.z + __logf(S.z));
    out.w = LOG_A - (M.w + __logf(S.w));
    *reinterpret_cast<float4*>(g + j) = out;
}

// ---------- T = exp(f + logK + g) with per-row block sums (deterministic) ----------
__global__ void t_kernel(const float* __restrict__ logK, const float* __restrict__ f,
                         const float* __restrict__ g, float* __restrict__ T,
                         float* __restrict__ bsum) {
    const int i = blockIdx.x, tid = threadIdx.x;
    const float fi = f[i];
    float local = 0.0f;
    for (int j = tid * 4; j < NT; j += 1024) {
        float4 v  = *reinterpret_cast<const float4*>(logK + (size_t)i * NT + j);
        float4 gg = *reinterpret_cast<const float4*>(g + j);
        float4 t;
        t.x = __expf(fi + v.x + gg.x);
        t.y = __expf(fi + v.y + gg.y);
        t.z = __expf(fi + v.z + gg.z);
        t.w = __expf(fi + v.w + gg.w);
        *reinterpret_cast<float4*>(T + (size_t)i * NT + j) = t;
        local += t.x + t.y + t.z + t.w;
    }
    __shared__ float red[256];
    red[tid] = local;
    __syncthreads();
    for (int w = 128; w > 0; w >>= 1) {
        if (tid < w) red[tid] += red[tid + w];
        __syncthreads();
    }
    if (tid == 0) bsum[i] = red[0];
}

__global__ void sum_kernel(const float* __restrict__ bsum, float* __restrict__ total) {
    const int tid = threadIdx.x;  // 1024
    float s = 0.0f;
    for (int i = tid; i < NA; i += 1024) s += bsum[i];
    __shared__ float red[1024];
    red[tid] = s;
    __syncthreads();
    for (int w = 512; w > 0; w >>= 1) {
        if (tid < w) red[tid] += red[tid + w];
        __syncthreads();
    }
    if (tid == 0) total[0] = red[0];
}

__global__ void scale_kernel(float* __restrict__ T, const float* __restrict__ total) {
    const float inv = 1.0f / (total[0] + 1e-8f);
    const size_t stride = (size_t)gridDim.x * 256 * 4;
    for (size_t p = ((size_t)blockIdx.x * 256 + threadIdx.x) * 4; p < (size_t)NA * NT; p += stride) {
        float4 t = *reinterpret_cast<float4*>(T + p);
        t.x *= inv; t.y *= inv; t.z *= inv; t.w *= inv;
        *reinterpret_cast<float4*>(T + p) = t;
    }
}

extern "C" void kernel_launch(void* const* d_in, const int* in_sizes, int n_in,
                              void* d_out, int out_size, void* d_ws, size_t ws_size,
                              hipStream_t stream) {
    const float* Aemb = (const float*)d_in[0];
    const float* Temb = (const float*)d_in[1];
    const float* Wq   = (const float*)d_in[2];
    const float* bq   = (const float*)d_in[3];
    const float* Wk   = (const float*)d_in[4];
    const float* bk   = (const float*)d_in[5];
    const float* W1   = (const float*)d_in[6];
    const float* b1   = (const float*)d_in[7];
    const float* W2   = (const float*)d_in[8];
    const float* b2   = (const float*)d_in[9];
    float* out = (float*)d_out;

    // workspace carve (floats, all offsets multiples of 64 -> 256B aligned)
    float* ws   = (float*)d_ws;
    float* q    = ws;            ws += (size_t)NA * DD;
    float* kbuf = ws;            ws += (size_t)NT * DD;
    float* qn   = ws;            ws += NA;
    float* kn   = ws;            ws += NT;
    float* h    = ws;            ws += (size_t)NA * DH;
    float* slog = ws;            ws += NA;
    float* logb = ws;            ws += NA;
    float* fv   = ws;            ws += NA;
    float* gv   = ws;            ws += NT;
    float* pm   = ws;            ws += (size_t)RC * NT;
    float* ps   = ws;            ws += (size_t)RC * NT;
    float* bsum = ws;            ws += NA;
    float* lse  = ws;            ws += 64;   // pad
    float* tot  = ws;            ws += 64;
    float* logK = ws;            // NA*NT floats (256 MB)

    // 1) projections q = A@Wq+bq, k = T@Wk+bk  (WMMA f32)
    wmma_gemm_bias<false><<<dim3(DD / 16, NA / 16), 32, 0, stream>>>(Aemb, Wq, bq, q, NA, DD, DD);
    wmma_gemm_bias<false><<<dim3(DD / 16, NT / 16), 32, 0, stream>>>(Temb, Wk, bk, kbuf, NT, DD, DD);
    rownorm_kernel<<<NA, 256, 0, stream>>>(q, qn);
    rownorm_kernel<<<NT, 256, 0, stream>>>(kbuf, kn);

    // 2) supply MLP -> log_b
    wmma_gemm_bias<true><<<dim3(DH / 16, NA / 16), 32, 0, stream>>>(Aemb, W1, b1, h, NA, DH, DD);
    logits_kernel<<<NA, 128, 0, stream>>>(h, W2, b2, slog);
    lse_kernel<<<1, 1024, 0, stream>>>(slog, lse);
    logb_kernel<<<NA / 256, 256, 0, stream>>>(slog, lse, logb);

    // 3) log_K = (qn + kn - 2 q.kT)/eps   (34 GFLOP WMMA GEMM, 16x64 per wave)
    logk_kernel<<<dim3(NT / 64, NA / 16), 32, 0, stream>>>(q, kbuf, qn, kn, logK);

    // 4) Sinkhorn iterations (HBM-bandwidth-bound streaming sweeps over logK)
    zero_kernel<<<NT / 256, 256, 0, stream>>>(gv);
    for (int it = 0; it < 50; ++it) {
        f_kernel<<<NA, 256, 0, stream>>>(logK, gv, logb, fv);
        g_partial_kernel<<<dim3(NT / 1024, RC), 256, 0, stream>>>(logK, fv, pm, ps);
        g_combine_kernel<<<NT / 1024, 256, 0, stream>>>(pm, ps, gv);
    }

    // 5) T = exp(f + logK + g), normalize by global sum (deterministic two-stage)
    t_kernel<<<NA, 256, 0, stream>>>(logK, fv, gv, out, bsum);
    sum_kernel<<<1, 1024, 0, stream>>>(bsum, tot);
    scale_kernel<<<16384, 256, 0, stream>>>(out, tot);
}